// RNNDisorderClassifier_general_14946486190685
// MI455X (gfx1250) — compile-verified
//
#include <hip/hip_runtime.h>
#include <hip/hip_bf16.h>
#include <math.h>

typedef __attribute__((ext_vector_type(2))) float v2f;
typedef __attribute__((ext_vector_type(8))) float v8f;

#define NHID 512
#define GATES 2048            // 4*NHID
#define BATCH 64
#define SEQ 512
#define NIN 128
#define LDS_STRIDE 260        // 256 cols + pad

// ---------------------------------------------------------------------------
// Setup kernels: f64 -> f32 conversion / K-paired transposed weight / zeroing
//
// WTP layout: rows of K-pairs. Element pair for (k, n), k even:
//   WTP[(k>>1) * 2*GATES + 2*n + 0] = W[k  ][n]
//   WTP[(k>>1) * 2*GATES + 2*n + 1] = W[k+1][n]
// where W[k][n] = (k < Kx) ? wih[n][k] : whh[n][k-Kx]
// => a lane's B fragment {B[kk][n], B[kk+1][n]} is one aligned b64 load.
// ---------------------------------------------------------------------------
__global__ void k_build_wt(const double* __restrict__ wih,
                           const double* __restrict__ whh,
                           float* __restrict__ WTP, int Kx, int Ktot) {
    long i = (long)blockIdx.x * blockDim.x + threadIdx.x;
    long total = (long)Ktot * GATES;
    if (i >= total) return;
    int kp   = (int)(i / (2 * GATES));     // K-pair row
    int rem  = (int)(i % (2 * GATES));
    int n    = rem >> 1;
    int half = rem & 1;
    int k    = 2 * kp + half;
    double v = (k < Kx) ? wih[(long)n * Kx + k]
                        : whh[(long)n * NHID + (k - Kx)];
    WTP[i] = (float)v;
}

__global__ void k_add_bias(const double* __restrict__ b1,
                           const double* __restrict__ b2,
                           float* __restrict__ out, int n) {
    int i = blockIdx.x * blockDim.x + threadIdx.x;
    if (i < n) out[i] = (float)(b1[i] + b2[i]);
}

__global__ void k_d2f(const double* __restrict__ in, float* __restrict__ out, long n) {
    long i = (long)blockIdx.x * blockDim.x + threadIdx.x;
    if (i < n) out[i] = (float)in[i];
}

__global__ void k_zero(float* __restrict__ p, long n) {
    long i = (long)blockIdx.x * blockDim.x + threadIdx.x;
    if (i < n) p[i] = 0.0f;
}

// ---------------------------------------------------------------------------
// One LSTM time step, fully fused:
//   gates = bias + x_t @ wih^T + h_prev @ whh^T   (via K-paired combined WTP)
//   i,f,g,o activations; c,h update; optional h -> sequence output
// grid = 32 WGs: blockIdx = mtile (4, rows of 16) x jtile (8, 64 hidden cols)
// block = 256 threads = 8 waves; each wave owns two 16x16 WMMA C tiles
// ---------------------------------------------------------------------------
__global__ __launch_bounds__(256) void lstm_step_kernel(
    const float* __restrict__ Ax,     // x_t base: element [m][k] at Ax + m*strideAm + k
    long strideAm,
    int Kx,                           // input-feature count (128 or 512), multiple of 4
    const float* __restrict__ WTP,    // (Kx+NHID)/2 x (2*GATES), K-paired [k/2][n][2]
    const float* __restrict__ bias,   // GATES (bih+bhh)
    const float* __restrict__ h_prev, // BATCH x NHID
    float* __restrict__ h_next,       // BATCH x NHID
    float* __restrict__ c_state,      // BATCH x NHID (owned per j-column -> in place)
    float* __restrict__ out_seq,      // element [m][j] at out_seq + m*strideOut + j; may be null
    long strideOut)
{
    __shared__ float gl[16 * LDS_STRIDE];

    const int tid   = threadIdx.x;
    const int lane  = tid & 31;
    const int wave  = tid >> 5;
    const int lmod  = lane & 15;
    const int lhalf = lane >> 4;          // 0: K{0,1}, 1: K{2,3} (A/B frag halves)
    const int m0    = (blockIdx.x & 3) * 16;
    const int j0    = (blockIdx.x >> 2) * 64;

    // Each wave handles columns [c0, c0+16) and [c0+16, c0+32) of the 256-wide
    // gate tile. Column c maps to gate slice c>>6, hidden col j0 + (c&63).
    const int cA = wave * 32;
    const int cB = cA + 16;
    const int nA = (cA >> 6) * NHID + j0 + (cA & 63) + lmod;   // global gate row (B col)
    const int nB = (cB >> 6) * NHID + j0 + (cB & 63) + lmod;

    const int arow = m0 + lmod;           // A-matrix row for this lane
    const int koff = lhalf * 2;           // lane K offset within a K=4 chunk

    v8f acc0 = {};
    v8f acc1 = {};

    // ---- K loop part 1: input contribution (A = x_t) ----
    {
        const float* ar = Ax + (long)arow * strideAm;
        for (int k = 0; k < Kx; k += 4) {
            const int kk = k + koff;                       // even
            v2f a = { ar[kk], ar[kk + 1] };                // b64
            const float* wp = WTP + (long)(kk >> 1) * (2 * GATES);
            v2f bA = *reinterpret_cast<const v2f*>(wp + 2 * nA);   // b64
            v2f bB = *reinterpret_cast<const v2f*>(wp + 2 * nB);   // b64
            acc0 = __builtin_amdgcn_wmma_f32_16x16x4_f32(false, a, false, bA,
                                                         (short)0, acc0, false, false);
            acc1 = __builtin_amdgcn_wmma_f32_16x16x4_f32(false, a, false, bB,
                                                         (short)0, acc1, false, false);
        }
    }
    // ---- K loop part 2: recurrent contribution (A = h_prev) ----
    {
        const float* hr   = h_prev + (long)arow * NHID;
        const float* WTPh = WTP + (long)(Kx >> 1) * (2 * GATES);
        for (int k = 0; k < NHID; k += 4) {
            const int kk = k + koff;                       // even
            v2f a = { hr[kk], hr[kk + 1] };                // b64
            const float* wp = WTPh + (long)(kk >> 1) * (2 * GATES);
            v2f bA = *reinterpret_cast<const v2f*>(wp + 2 * nA);   // b64
            v2f bB = *reinterpret_cast<const v2f*>(wp + 2 * nB);   // b64
            acc0 = __builtin_amdgcn_wmma_f32_16x16x4_f32(false, a, false, bA,
                                                         (short)0, acc0, false, false);
            acc1 = __builtin_amdgcn_wmma_f32_16x16x4_f32(false, a, false, bB,
                                                         (short)0, acc1, false, false);
        }
    }

    // ---- stage raw gate values in LDS (C layout: VGPR v -> rows v, v+8) ----
    const int rbase = lhalf * 8;
#pragma unroll
    for (int v = 0; v < 8; ++v) {
        gl[(v + rbase) * LDS_STRIDE + cA + lmod] = acc0[v];
        gl[(v + rbase) * LDS_STRIDE + cB + lmod] = acc1[v];
    }
    __syncthreads();

    // ---- activations + state update: 16 rows x 64 hidden cols ----
    for (int e = tid; e < 16 * 64; e += 256) {
        const int r  = e >> 6;
        const int jj = e & 63;
        const int m  = m0 + r;
        const int jg = j0 + jj;
        const float xi = gl[r * LDS_STRIDE +        jj] + bias[           jg];
        const float xf = gl[r * LDS_STRIDE +  64 + jj] + bias[    NHID + jg];
        const float xg = gl[r * LDS_STRIDE + 128 + jj] + bias[2 * NHID + jg];
        const float xo = gl[r * LDS_STRIDE + 192 + jj] + bias[3 * NHID + jg];
        const float ig = 1.0f / (1.0f + expf(-xi));
        const float fg = 1.0f / (1.0f + expf(-xf));
        const float gg = tanhf(xg);
        const float og = 1.0f / (1.0f + expf(-xo));
        const long idx = (long)m * NHID + jg;
        const float c  = fg * c_state[idx] + ig * gg;
        const float h  = og * tanhf(c);
        c_state[idx] = c;
        h_next[idx]  = h;
        if (out_seq) out_seq[(long)m * strideOut + jg] = h;
    }
}

// ---------------------------------------------------------------------------
// FC head: hidden = relu(h @ fc0_w^T + fc0_b); out = sigmoid(hidden @ fc1_w^T + fc1_b)
// one workgroup per batch row
// ---------------------------------------------------------------------------
__global__ __launch_bounds__(256) void fc_head_kernel(
    const float* __restrict__ h,      // BATCH x NHID
    const float* __restrict__ w0,     // NHID x NHID row-major [n][k]
    const float* __restrict__ b0,     // NHID
    const float* __restrict__ w1,     // NHID
    const float* __restrict__ b1,     // 1
    float* __restrict__ out)          // BATCH
{
    __shared__ float hid[NHID];
    __shared__ float red[256];
    const int m = blockIdx.x;
    const float* hm = h + (long)m * NHID;

    for (int n = threadIdx.x; n < NHID; n += 256) {
        float s = b0[n];
        const float* wr = w0 + (long)n * NHID;
        for (int k = 0; k < NHID; ++k) s += hm[k] * wr[k];
        hid[n] = fmaxf(s, 0.0f);
    }
    __syncthreads();

    float s = 0.0f;
    for (int k = threadIdx.x; k < NHID; k += 256) s += hid[k] * w1[k];
    red[threadIdx.x] = s;
    __syncthreads();
    for (int off = 128; off > 0; off >>= 1) {
        if (threadIdx.x < off) red[threadIdx.x] += red[threadIdx.x + off];
        __syncthreads();
    }
    if (threadIdx.x == 0) out[m] = 1.0f / (1.0f + expf(-(red[0] + b1[0])));
}

// ---------------------------------------------------------------------------
extern "C" void kernel_launch(void* const* d_in, const int* in_sizes, int n_in,
                              void* d_out, int out_size, void* d_ws, size_t ws_size,
                              hipStream_t stream) {
    const float*  x     = (const float*) d_in[0];
    const double* wih0  = (const double*)d_in[1];
    const double* whh0  = (const double*)d_in[2];
    const double* bih0  = (const double*)d_in[3];
    const double* bhh0  = (const double*)d_in[4];
    const double* wih1  = (const double*)d_in[5];
    const double* whh1  = (const double*)d_in[6];
    const double* bih1  = (const double*)d_in[7];
    const double* bhh1  = (const double*)d_in[8];
    const double* fc0_w = (const double*)d_in[9];
    const double* fc0_b = (const double*)d_in[10];
    const double* fc1_w = (const double*)d_in[11];
    const double* fc1_b = (const double*)d_in[12];

    // ---- workspace layout (floats; all offsets 8-byte aligned) ----
    float* ws = (float*)d_ws;
    float* WT0   = ws;                          // 640 * 2048 (K-paired)
    float* WT1   = WT0 + 640L * GATES;          // 1024 * 2048 (K-paired)
    float* bias0 = WT1 + 1024L * GATES;         // 2048
    float* bias1 = bias0 + GATES;               // 2048
    float* fc0wf = bias1 + GATES;               // 512*512
    float* fc0bf = fc0wf + (long)NHID * NHID;   // 512
    float* fc1wf = fc0bf + NHID;                // 512
    float* fc1bf = fc1wf + NHID;                // 1 (+pad)
    float* hA    = fc1bf + 16;                  // 64*512  (h double buffer 0)
    float* hB    = hA + (long)BATCH * NHID;     // 64*512  (h double buffer 1)
    float* cS    = hB + (long)BATCH * NHID;     // 64*512  (cell state)
    float* hseq  = cS + (long)BATCH * NHID;     // 64*512*512 (layer-0 output sequence)

    // ---- setup: weight conversion / transposition ----
    {
        long n0 = 640L * GATES;
        k_build_wt<<<(unsigned)((n0 + 255) / 256), 256, 0, stream>>>(wih0, whh0, WT0, NIN, NIN + NHID);
        long n1 = 1024L * GATES;
        k_build_wt<<<(unsigned)((n1 + 255) / 256), 256, 0, stream>>>(wih1, whh1, WT1, NHID, NHID + NHID);
        k_add_bias<<<(GATES + 255) / 256, 256, 0, stream>>>(bih0, bhh0, bias0, GATES);
        k_add_bias<<<(GATES + 255) / 256, 256, 0, stream>>>(bih1, bhh1, bias1, GATES);
        long nf = (long)NHID * NHID;
        k_d2f<<<(unsigned)((nf + 255) / 256), 256, 0, stream>>>(fc0_w, fc0wf, nf);
        k_d2f<<<(NHID + 255) / 256, 256, 0, stream>>>(fc0_b, fc0bf, NHID);
        k_d2f<<<(NHID + 255) / 256, 256, 0, stream>>>(fc1_w, fc1wf, NHID);
        k_d2f<<<1, 256, 0, stream>>>(fc1_b, fc1bf, 1);
        // zero h double buffers + cell state (contiguous)
        long nz = 3L * BATCH * NHID;
        k_zero<<<(unsigned)((nz + 255) / 256), 256, 0, stream>>>(hA, nz);
    }

    // ---- layer 0 recurrence: gates = bias0 + x_t@wih0^T + h@whh0^T ----
    for (int t = 0; t < SEQ; ++t) {
        const float* hp = (t & 1) ? hB : hA;
        float*       hn = (t & 1) ? hA : hB;
        lstm_step_kernel<<<32, 256, 0, stream>>>(
            x + (long)t * NIN, (long)SEQ * NIN, NIN,
            WT0, bias0, hp, hn, cS,
            hseq + (long)t * NHID, (long)SEQ * NHID);
    }

    // ---- reset states for layer 1 ----
    {
        long nz = 3L * BATCH * NHID;
        k_zero<<<(unsigned)((nz + 255) / 256), 256, 0, stream>>>(hA, nz);
    }

    // ---- layer 1 recurrence: input = hseq ----
    for (int t = 0; t < SEQ; ++t) {
        const float* hp = (t & 1) ? hB : hA;
        float*       hn = (t & 1) ? hA : hB;
        lstm_step_kernel<<<32, 256, 0, stream>>>(
            hseq + (long)t * NHID, (long)SEQ * NHID, NHID,
            WT1, bias1, hp, hn, cS,
            (float*)nullptr, 0L);
    }
    // after t=511 (odd), h_next == hA holds h2[:, -1]

    // ---- FC head ----
    fc_head_kernel<<<BATCH, 256, 0, stream>>>(hA, fc0wf, fc0bf, fc1wf, fc1bf, (float*)d_out);
}